// AELossV2_6562710028536
// MI455X (gfx1250) — compile-verified
//
#include <hip/hip_runtime.h>

// AELoss: pull = weighted-mean BCEWithLogits; push = masked mean of pairwise
// sigmoid distances over [B=8, N=2048, N=2048]. Memory-bound on the 33.5 MB
// bool mask; everything else is recomputed from LDS-resident s tiles.

#define BB 8
#define NN 2048
#define THR 0.6f   // 0.5 + MARGIN_PUSH

typedef float v2f  __attribute__((ext_vector_type(2)));
typedef float v8f  __attribute__((ext_vector_type(8)));
typedef unsigned u32x2 __attribute__((ext_vector_type(2)));

// ---------------------------------------------------------------------------
// Wave32 sum reduction via V_WMMA_F32_16X16X4_F32 with B = ones:
//   A vgpr0 = per-lane value v (element (M = lane&15, K = lane<16 ? 0 : 2)),
//   A vgpr1 = 0  ->  D[m,n] = v[m] + v[m+16] for all n.
// Per-lane sum of the 8 D vgprs covers M=0..7 (lanes 0-15) or M=8..15
// (lanes 16-31); adding the lane^16 partner yields the full 32-lane sum.
// ---------------------------------------------------------------------------
__device__ __forceinline__ float wave_sum_wmma(float v) {
    v2f a; a[0] = v;    a[1] = 0.0f;
    v2f b; b[0] = 1.0f; b[1] = 1.0f;
    v8f c = {0.f, 0.f, 0.f, 0.f, 0.f, 0.f, 0.f, 0.f};
    v8f d = __builtin_amdgcn_wmma_f32_16x16x4_f32(
        /*neg_a=*/false, a, /*neg_b=*/false, b,
        /*c_mod=*/(short)0, c, /*reuse_a=*/false, /*reuse_b=*/false);
    float t = d[0] + d[1] + d[2] + d[3] + d[4] + d[5] + d[6] + d[7];
    t += __shfl_xor(t, 16, 32);
    return t;
}

// Block (256 threads = 8 waves) sum. `red` is 8 floats of shared memory.
// Must be called by all threads with EXEC all-1s (WMMA requirement).
__device__ __forceinline__ float block_sum(float v, float* red) {
    float t = wave_sum_wmma(v);
    const int wid  = threadIdx.x >> 5;
    const int lane = threadIdx.x & 31;
    if (lane == 0) red[wid] = t;
    __syncthreads();
    float total = 0.f;
#pragma unroll
    for (int w = 0; w < 8; ++w) total += red[w];
    __syncthreads();
    return total;
}

// ---------------------------------------------------------------------------
// Kernels
// ---------------------------------------------------------------------------
__global__ void ae_init_kernel(float* acc) {
    if (threadIdx.x < 4) acc[threadIdx.x] = 0.0f;  // [0]=pull_num [1]=pull_den [2]=push_sum [3]=cnt bits
}

// Pull term + materialize s = sigmoid(lof_tag_avg_img) into workspace.
__global__ void ae_pull_sig_kernel(const float* __restrict__ tag_in,
                                   const float* __restrict__ avg_in,
                                   const float* __restrict__ gath_in,
                                   const float* __restrict__ cent_in,
                                   float* __restrict__ s_out,
                                   float* __restrict__ acc) {
    __shared__ float red[8];
    const int idx = blockIdx.x * blockDim.x + threadIdx.x;  // grid exactly covers B*N
    const float x  = tag_in[idx];
    const float g  = gath_in[idx];
    const float cw = cent_in[idx];
    // round(sigmoid(g)) == (g > 0)  (g==0 -> 0.5 -> round-half-even -> 0)
    const float target = (g > 0.0f) ? 1.0f : 0.0f;
    // stable softplus
    const float sp  = fmaxf(x, 0.0f) + log1pf(expf(-fabsf(x)));
    const float tag = sp - x * target;
    s_out[idx] = 1.0f / (1.0f + expf(-avg_in[idx]));

    const float num = block_sum(tag * cw, red);
    const float den = block_sum(cw, red);
    if (threadIdx.x == 0) {
        atomicAdd(&acc[0], num);
        atomicAdd(&acc[1], den);
    }
}

// Push term. Tile: 16 rows (i) x 128 cols (j) x 8 batches per block.
// tx = tid&15 handles 8 consecutive j via one 8-byte mask load (full 128B
// cacheline per (i,b) row across 16 threads). ty = tid>>4 is the i row.
__global__ void ae_push_kernel(const float* __restrict__ s,
                               const unsigned char* __restrict__ mask,
                               float* __restrict__ acc_push,
                               unsigned* __restrict__ acc_cnt) {
    __shared__ float sj[BB][128];
    __shared__ float si[BB][16];
    __shared__ float red[8];
    const int tid = threadIdx.x;
    const int tx  = tid & 15;
    const int ty  = tid >> 4;
    const int j0  = blockIdx.x * 128;
    const int i0  = blockIdx.y * 16;

    // stage s tiles into LDS
#pragma unroll
    for (int k = 0; k < 4; ++k) {
        const int idx = tid + 256 * k;         // 1024 = 8*128
        const int b = idx >> 7, jj = idx & 127;
        sj[b][jj] = s[b * NN + j0 + jj];
    }
    if (tid < 128) {
        const int b = tid >> 4, ii = tid & 15;
        si[b][ii] = s[b * NN + i0 + ii];
    }
    __syncthreads();

    float myi[BB];
#pragma unroll
    for (int b = 0; b < BB; ++b) myi[b] = si[b][ty];

    const int jbase = tx * 8;
    // dist_mask[i,j] = OR over b of (s[b,i] != s[b,j]); 8-bit mask over my 8 j's
    unsigned diffm = 0;
#pragma unroll
    for (int b = 0; b < BB; ++b) {
#pragma unroll
        for (int q = 0; q < 8; ++q) {
            diffm |= (sj[b][jbase + q] != myi[b]) ? (1u << q) : 0u;
        }
    }

    const size_t rowbase = (size_t)(i0 + ty) * NN + (size_t)(j0 + jbase);
    float psum = 0.f;
    unsigned cnt = 0;
#pragma unroll
    for (int b = 0; b < BB; ++b) {
        const unsigned char* p = mask + (size_t)b * (size_t)NN * (size_t)NN + rowbase;
        if (b + 1 < BB) __builtin_prefetch(p + (size_t)NN * (size_t)NN, 0, 1);
        const u32x2 mv = __builtin_nontemporal_load((const u32x2*)p);  // 8 mask bytes, read-once
#pragma unroll
        for (int q = 0; q < 8; ++q) {
            const unsigned byte = (q < 4) ? ((mv[0] >> (8 * q)) & 0xFFu)
                                          : ((mv[1] >> (8 * (q - 4))) & 0xFFu);
            const unsigned sel = (byte != 0u) & ((diffm >> q) & 1u);
            const float d = fabsf(sj[b][jbase + q] - myi[b]);
            psum = fmaf((float)sel, THR - d, psum);   // branchless accumulate
            cnt += sel;
        }
    }

    const float tot_push = block_sum(psum, red);
    const float tot_cnt  = block_sum((float)cnt, red);  // per-block cnt <= 16384: exact in f32
    if (tid == 0) {
        atomicAdd(acc_push, tot_push);
        atomicAdd(acc_cnt, (unsigned)(tot_cnt + 0.5f));
    }
}

__global__ void ae_final_kernel(const float* __restrict__ acc,
                                const unsigned* __restrict__ cnt,
                                float* __restrict__ out) {
    out[0] = acc[0] / acc[1];                       // PULL_FACTOR = 1
    const unsigned c = *cnt;
    out[1] = (c > 0u) ? (acc[2] / (float)c) : 0.0f; // PUSH_FACTOR = 1
}

// ---------------------------------------------------------------------------
// Launch
// ---------------------------------------------------------------------------
extern "C" void kernel_launch(void* const* d_in, const int* in_sizes, int n_in,
                              void* d_out, int out_size, void* d_ws, size_t ws_size,
                              hipStream_t stream) {
    (void)in_sizes; (void)n_in; (void)out_size; (void)ws_size;
    const float*         tag  = (const float*)d_in[0];         // lof_tag_img        [B,N]
    const float*         avg  = (const float*)d_in[1];         // lof_tag_avg_img    [B,N]
    const float*         gath = (const float*)d_in[2];         // lof_tag_avg_gather [B,N]
    const unsigned char* mask = (const unsigned char*)d_in[3]; // mask (bool)        [B,N,N]
    const float*         cent = (const float*)d_in[4];         // centerness_img     [B,N]
    float* out = (float*)d_out;

    float*    acc  = (float*)d_ws;          // acc[0..2] floats, acc[3] bits = unsigned count
    unsigned* cntp = (unsigned*)(acc + 3);
    float*    sbuf = acc + 16;              // s = sigmoid(avg), B*N floats

    hipLaunchKernelGGL(ae_init_kernel, dim3(1), dim3(32), 0, stream, acc);
    hipLaunchKernelGGL(ae_pull_sig_kernel, dim3((BB * NN) / 256), dim3(256), 0, stream,
                       tag, avg, gath, cent, sbuf, acc);
    hipLaunchKernelGGL(ae_push_kernel, dim3(NN / 128, NN / 16), dim3(256), 0, stream,
                       sbuf, mask, acc + 2, cntp);
    hipLaunchKernelGGL(ae_final_kernel, dim3(1), dim3(1), 0, stream, acc, cntp, out);
}